// TotalLoss_3109556322827
// MI455X (gfx1250) — compile-verified
//
#include <hip/hip_runtime.h>
#include <hip/hip_bf16.h>
#include <math.h>

typedef __attribute__((ext_vector_type(2))) float v2f;
typedef __attribute__((ext_vector_type(4))) float f4;
typedef __attribute__((ext_vector_type(8))) float v8f;

#define P_PARTS 6
#define NUM_CLASSES 50000
#define BATCH 128
#define PART_DIM 256
#define FEAT_DIM (P_PARTS * PART_DIM)   // 1536
#define NROWS (P_PARTS * BATCH)         // 768
#define NVEC (NUM_CLASSES / 4)          // 12500

// ---------------------------------------------------------------------------
// Kernel 1: fused single-pass row statistics over id_logits + soft_labels.
// One block per (part, batch) row; 256 threads stream 50000 floats of each
// tensor with float4 non-temporal loads, maintaining an online softmax at
// two temperatures plus the linear sums needed by CE / KL.
// Transcendental budget: 1 exp + 1 log per element (exp(d) = exp(d/3)^3).
// ---------------------------------------------------------------------------
__global__ __launch_bounds__(256) void row_stats_kernel(
    const float* __restrict__ logits, const float* __restrict__ soft,
    const int* __restrict__ labels, float* __restrict__ out /* NROWS*8 */) {
  const int row = blockIdx.x;          // p*BATCH + b
  const int b = row & (BATCH - 1);
  const int tid = threadIdx.x;
  const float* x = logits + (size_t)row * NUM_CLASSES;
  const float* q = soft + (size_t)row * NUM_CLASSES;

  float m = -3.402823466e38f;
  float s1 = 0.f, s3 = 0.f, sx = 0.f, qx = 0.f, qlq = 0.f;
  // Label logit gathered once per row (not in the 50k-element hot loop).
  float xl = (tid == 0) ? x[labels[b]] : 0.f;
  const float INV_T = 1.0f / 3.0f;

  for (int v = tid; v < NVEC; v += 256) {
    f4 xv = __builtin_nontemporal_load((const f4*)x + v);
    f4 qv = __builtin_nontemporal_load((const f4*)q + v);
#pragma unroll
    for (int e = 0; e < 4; ++e) {
      float xe = xv[e];
      float qe = qv[e];
      sx += xe;
      qx += qe * xe;
      qlq += qe * __logf(qe);
      float d = xe - m;
      if (d > 0.f) {            // rare after warmup (~ln(n) times per thread)
        float e3 = __expf(-d * INV_T);
        float e1 = e3 * e3 * e3;   // exp(-d)
        s1 = s1 * e1 + 1.0f;
        s3 = s3 * e3 + 1.0f;
        m = xe;
      } else {
        float e3 = __expf(d * INV_T);
        s3 += e3;
        s1 += e3 * e3 * e3;        // exp(d)
      }
    }
  }

  __shared__ float Sm[256], S1[256], S3[256], Sx[256], Xl[256], Qx[256], Ql[256];
  Sm[tid] = m; S1[tid] = s1; S3[tid] = s3; Sx[tid] = sx;
  Xl[tid] = xl; Qx[tid] = qx; Ql[tid] = qlq;
  __syncthreads();
  for (int off = 128; off > 0; off >>= 1) {
    if (tid < off) {
      float ma = Sm[tid], mb = Sm[tid + off];
      float M = fmaxf(ma, mb);
      float ea3 = __expf((ma - M) * INV_T);
      float eb3 = __expf((mb - M) * INV_T);
      S1[tid] = S1[tid] * (ea3 * ea3 * ea3) + S1[tid + off] * (eb3 * eb3 * eb3);
      S3[tid] = S3[tid] * ea3 + S3[tid + off] * eb3;
      Sm[tid] = M;
      Sx[tid] += Sx[tid + off];
      Xl[tid] += Xl[tid + off];
      Qx[tid] += Qx[tid + off];
      Ql[tid] += Ql[tid + off];
    }
    __syncthreads();
  }
  if (tid == 0) {
    float* o = out + row * 8;
    o[0] = Sm[0]; o[1] = S1[0]; o[2] = S3[0]; o[3] = Sx[0];
    o[4] = Xl[0]; o[5] = Qx[0]; o[6] = Ql[0]; o[7] = 0.f;
  }
}

// ---------------------------------------------------------------------------
// Kernel 2: concatenate the 6 part features per sample and L2-normalize the
// resulting [128, 1536] rows (both RGB and gray feature stacks).
// ---------------------------------------------------------------------------
__global__ __launch_bounds__(256) void normalize_kernel(
    const float* __restrict__ idf, const float* __restrict__ gry,
    float* __restrict__ featN, float* __restrict__ grayN) {
  const int b = blockIdx.x;      // sample
  const int tid = threadIdx.x;   // part dim index (0..255)
  float v[P_PARTS], g[P_PARTS];
  float ss = 0.f, sg = 0.f;
#pragma unroll
  for (int i = 0; i < P_PARTS; ++i) {
    float a = idf[i * BATCH * PART_DIM + b * PART_DIM + tid];
    float c = gry[i * BATCH * PART_DIM + b * PART_DIM + tid];
    v[i] = a; g[i] = c;
    ss += a * a; sg += c * c;
  }
  __shared__ float A[256], B[256];
  A[tid] = ss; B[tid] = sg;
  __syncthreads();
  for (int off = 128; off > 0; off >>= 1) {
    if (tid < off) { A[tid] += A[tid + off]; B[tid] += B[tid + off]; }
    __syncthreads();
  }
  const float i1 = 1.0f / sqrtf(A[0]);
  const float i2 = 1.0f / sqrtf(B[0]);
#pragma unroll
  for (int i = 0; i < P_PARTS; ++i) {
    featN[b * FEAT_DIM + i * PART_DIM + tid] = v[i] * i1;
    grayN[b * FEAT_DIM + i * PART_DIM + tid] = g[i] * i2;
  }
}

// ---------------------------------------------------------------------------
// Kernel 3: Gram matrices via V_WMMA_F32_16X16X4_F32 (full f32 precision).
// One wave per 16x16 output tile; two accumulators (feat.featT, feat.grayT)
// share the A operand. Unit-norm rows => dist = sqrt(max(2-2g,0)+1e-12).
// ---------------------------------------------------------------------------
__global__ __launch_bounds__(32) void gram_wmma_kernel(
    const float* __restrict__ featN, const float* __restrict__ grayN,
    float* __restrict__ dist1, float* __restrict__ dist2) {
  const int ti = blockIdx.x >> 3;     // row tile 0..7
  const int tj = blockIdx.x & 7;      // col tile 0..7
  const int l = threadIdx.x;
  const int lm = l & 15;
  const int hi = l >> 4;              // 0: K=0,1  1: K=2,3 (A 16x4 f32 layout)
  const int m0 = ti * 16, n0 = tj * 16;

  const float* arow = featN + (m0 + lm) * FEAT_DIM + 2 * hi;
  const float* brow = featN + (n0 + lm) * FEAT_DIM + 2 * hi;  // B = featN^T
  const float* crow = grayN + (n0 + lm) * FEAT_DIM + 2 * hi;  // B = grayN^T

  v8f acc1 = {0.f, 0.f, 0.f, 0.f, 0.f, 0.f, 0.f, 0.f};
  v8f acc2 = {0.f, 0.f, 0.f, 0.f, 0.f, 0.f, 0.f, 0.f};

  for (int k0 = 0; k0 < FEAT_DIM; k0 += 4) {
    v2f a  = *(const v2f*)(arow + k0);
    v2f b1 = *(const v2f*)(brow + k0);
    v2f b2 = *(const v2f*)(crow + k0);
    acc1 = __builtin_amdgcn_wmma_f32_16x16x4_f32(false, a, false, b1,
                                                 (short)0, acc1, false, false);
    acc2 = __builtin_amdgcn_wmma_f32_16x16x4_f32(false, a, false, b2,
                                                 (short)0, acc2, false, false);
  }

#pragma unroll
  for (int r = 0; r < 8; ++r) {
    const int row = m0 + r + hi * 8;   // C/D layout: VGPR r -> M=r / M=r+8
    const int col = n0 + lm;
    dist1[row * BATCH + col] = sqrtf(fmaxf(2.f - 2.f * acc1[r], 0.f) + 1e-12f);
    dist2[row * BATCH + col] = sqrtf(fmaxf(2.f - 2.f * acc2[r], 0.f) + 1e-12f);
  }
}

// ---------------------------------------------------------------------------
// Kernel 4: batch-hard mining per anchor row for both triplet terms.
// ---------------------------------------------------------------------------
__global__ __launch_bounds__(128) void tri_mine_kernel(
    const float* __restrict__ dist1, const float* __restrict__ dist2,
    const int* __restrict__ labels, float* __restrict__ terms /* 2*BATCH */) {
  const int i = threadIdx.x;
  __shared__ int lab[BATCH];
  lab[i] = labels[i];
  __syncthreads();
  const int li = lab[i];
  float ap1 = -1e30f, an1 = 1e30f, ap2 = -1e30f, an2 = 1e30f;
  bool hp1 = false, hp2 = false, hn = false;
  for (int j = 0; j < BATCH; ++j) {
    const bool eq = (lab[j] == li);
    const float d1 = dist1[i * BATCH + j];
    const float d2 = dist2[i * BATCH + j];
    if (eq) {
      if (j != i) { hp1 = true; ap1 = fmaxf(ap1, d1); }
      hp2 = true; ap2 = fmaxf(ap2, d2);
    } else {
      hn = true;
      an1 = fminf(an1, d1);
      an2 = fminf(an2, d2);
    }
  }
  const float a1 = hp1 ? ap1 : 0.f, n1 = hn ? an1 : 1e6f;
  const float a2 = hp2 ? ap2 : 0.f, n2 = hn ? an2 : 1e6f;
  terms[i]         = fmaxf(a1 - n1 + 0.3f, 0.f);
  terms[BATCH + i] = fmaxf(a2 - n2 + 0.3f, 0.f);
}

// ---------------------------------------------------------------------------
// Kernel 5: final combine -> scalar total loss.
// ---------------------------------------------------------------------------
__global__ __launch_bounds__(256) void final_kernel(
    const float* __restrict__ rowstats, const float* __restrict__ entw,
    const float* __restrict__ adv, const int* __restrict__ modality,
    const int* __restrict__ epoch, const float* __restrict__ terms,
    float* __restrict__ out) {
  const int tid = threadIdx.x;
  const float INV_T = 1.0f / 3.0f;
  float aid = 0.f, akl = 0.f, aadv = 0.f, at1 = 0.f, at2 = 0.f;

  for (int r = tid; r < NROWS; r += 256) {
    const float* o = rowstats + r * 8;
    const float m = o[0], s1 = o[1], s3 = o[2], sx = o[3];
    const float xl = o[4], qx = o[5], qlq = o[6];
    const float L = logf(s1);
    // label-smoothed CE: (1-s)*nll + s*smooth
    aid += m + L - 0.9f * xl - 0.1f * (sx / (float)NUM_CLASSES);
    // KL(q || softmax(x/T))
    float kl = qlq - qx * INV_T + m * INV_T + logf(s3);
    kl = fminf(kl, 5.0f);
    akl += kl * entw[r];
    // adversarial CE over 2 classes
    const float a0 = adv[r * 2], a1 = adv[r * 2 + 1];
    const int sel = modality[r & (BATCH - 1)];
    const float mm = fmaxf(a0, a1);
    const float lse = mm + logf(__expf(a0 - mm) + __expf(a1 - mm));
    aadv += lse - (sel == 0 ? a0 : a1);
  }
  if (tid < BATCH) { at1 = terms[tid]; at2 = terms[BATCH + tid]; }

  __shared__ float R0[256], R1[256], R2[256], R3[256], R4[256];
  R0[tid] = aid; R1[tid] = akl; R2[tid] = aadv; R3[tid] = at1; R4[tid] = at2;
  __syncthreads();
  for (int off = 128; off > 0; off >>= 1) {
    if (tid < off) {
      R0[tid] += R0[tid + off]; R1[tid] += R1[tid + off];
      R2[tid] += R2[tid + off]; R3[tid] += R3[tid + off];
      R4[tid] += R4[tid + off];
    }
    __syncthreads();
  }
  if (tid == 0) {
    const float L_id = R0[0] / (float)NROWS;
    const float L_tri = R3[0] / (float)BATCH + 0.5f * (R4[0] / (float)BATCH);
    // sum_p mean_b(kl*w) * T^2 / P  ==  (sum_all / BATCH) * 9/6
    const float L_graph = (epoch[0] >= 20) ? (R1[0] / (float)BATCH) * 1.5f : 0.f;
    const float L_adv = R2[0] / (float)NROWS;
    out[0] = L_id + 1.0f * L_tri + 0.1f * L_graph + 0.1f * L_adv;
  }
}

// ---------------------------------------------------------------------------
extern "C" void kernel_launch(void* const* d_in, const int* in_sizes, int n_in,
                              void* d_out, int out_size, void* d_ws, size_t ws_size,
                              hipStream_t stream) {
  const float* id_logits     = (const float*)d_in[0];
  const float* id_features   = (const float*)d_in[1];
  const float* gray_features = (const float*)d_in[2];
  const float* soft_labels   = (const float*)d_in[3];
  const float* entw          = (const float*)d_in[4];
  const float* adv           = (const float*)d_in[5];
  const int*   labels        = (const int*)d_in[6];
  const int*   modality      = (const int*)d_in[7];
  const int*   epoch         = (const int*)d_in[8];
  float* out = (float*)d_out;

  float* ws       = (float*)d_ws;
  float* rowstats = ws;                                   // 768*8   = 6144
  float* featN    = rowstats + NROWS * 8;                 // 128*1536
  float* grayN    = featN + BATCH * FEAT_DIM;             // 128*1536
  float* dist1    = grayN + BATCH * FEAT_DIM;             // 128*128
  float* dist2    = dist1 + BATCH * BATCH;                // 128*128
  float* terms    = dist2 + BATCH * BATCH;                // 256

  normalize_kernel<<<BATCH, 256, 0, stream>>>(id_features, gray_features,
                                              featN, grayN);
  gram_wmma_kernel<<<64, 32, 0, stream>>>(featN, grayN, dist1, dist2);
  tri_mine_kernel<<<1, BATCH, 0, stream>>>(dist1, dist2, labels, terms);
  row_stats_kernel<<<NROWS, 256, 0, stream>>>(id_logits, soft_labels, labels,
                                              rowstats);
  final_kernel<<<1, 256, 0, stream>>>(rowstats, entw, adv, modality, epoch,
                                      terms, out);
}